// GarNet_13950053778179
// MI455X (gfx1250) — compile-verified
//
#include <hip/hip_runtime.h>
#include <math.h>

// Problem constants (match reference)
#define BQ  64
#define VQ  4096
#define FQ  16
#define AQ  8
#define PQ  16
#define NFQ 16

typedef __attribute__((ext_vector_type(2))) float v2f;
typedef __attribute__((ext_vector_type(8))) float v8f;

// D = A(16x4 f32) * B(4x16 f32) + C(16x16 f32)
__device__ __forceinline__ v8f wmma4(v2f a, v2f b, v8f c) {
  return __builtin_amdgcn_wmma_f32_16x16x4_f32(
      /*neg_a=*/false, a, /*neg_b=*/false, b,
      /*c_mod=*/(short)0, c, /*reuse_a=*/false, /*reuse_b=*/false);
}

__device__ __forceinline__ v8f vzero8() {
  v8f z;
#pragma unroll
  for (int i = 0; i < 8; ++i) z[i] = 0.0f;
  return z;
}

// ---------------------------------------------------------------------------
// Kernel 0: zero the global agg accumulator (workspace is poisoned / reused)
// ---------------------------------------------------------------------------
__global__ __launch_bounds__(256) void garnet_zero(float* __restrict__ p, int n) {
  int i = blockIdx.x * 256 + threadIdx.x;
  if (i < n) p[i] = 0.0f;
}

// ---------------------------------------------------------------------------
// Kernel 1: per (batch, vertex-chunk) partial aggregation
//   agg[b,a,p] += sum_v ew[b,v,a] * feat[b,v,p]
// grid = (B, V/256), block = 256 (8 waves, each wave does 2 vertex tiles of 16)
// ---------------------------------------------------------------------------
__global__ __launch_bounds__(256) void garnet_agg(
    const float* __restrict__ data, const int* __restrict__ num_vertex,
    const float* __restrict__ W_flr, const float* __restrict__ b_flr,
    const float* __restrict__ W_s,  const float* __restrict__ b_s,
    float* __restrict__ agg_g)
{
  const int b    = blockIdx.x;
  const int lane = threadIdx.x & 31;
  const int wave = threadIdx.x >> 5;   // 0..7
  const int r    = lane & 15;          // row / col / lane index within half
  const int hi   = lane >> 4;          // which K half-pair

  __shared__ float ew_lds[8][16][17];
  __shared__ float ft_lds[8][16][17];
  __shared__ float agg_lds[16][16];

  agg_lds[threadIdx.x >> 4][threadIdx.x & 15] = 0.0f;  // 256 threads == 256 slots
  __syncthreads();

  const int nv = num_vertex[b];

  // Weight fragments in WMMA B-layout, K chunks of 4: B[k,n], n = r, k = kc*4+2*hi+{0,1}
  v2f wflr[4], ws[4];
#pragma unroll
  for (int kc = 0; kc < 4; ++kc) {
    const int k0 = kc * 4 + hi * 2;
    wflr[kc].x = W_flr[(k0 + 0) * PQ + r];
    wflr[kc].y = W_flr[(k0 + 1) * PQ + r];
    ws[kc].x = (r < AQ) ? W_s[(k0 + 0) * AQ + r] : 0.0f;   // pad A-dim 8 -> 16
    ws[kc].y = (r < AQ) ? W_s[(k0 + 1) * AQ + r] : 0.0f;
  }
  const float bflr_n = b_flr[r];
  const float bs_n   = (r < AQ) ? b_s[r] : 0.0f;

  v8f agg = vzero8();  // persistent accumulator (C layout: M=a, N=p)

  const float* dbase = data + (size_t)b * VQ * FQ;

#pragma unroll
  for (int tt = 0; tt < 2; ++tt) {
    const int t  = blockIdx.y * 16 + wave + tt * 8;  // tile of 16 vertices
    const int v0 = t * 16;

    // data tile in A-layout: A[m,k], m = r, k = kc*4+2*hi+{0,1}
    const float* drow = dbase + (size_t)(v0 + r) * FQ;
    v2f af[4];
#pragma unroll
    for (int kc = 0; kc < 4; ++kc) {
      const int k0 = kc * 4 + hi * 2;
      af[kc].x = drow[k0 + 0];
      af[kc].y = drow[k0 + 1];
    }

    v8f feat = vzero8();
    v8f dist = vzero8();
#pragma unroll
    for (int kc = 0; kc < 4; ++kc) {
      feat = wmma4(af[kc], wflr[kc], feat);   // [16v x 16p]
      dist = wmma4(af[kc], ws[kc],   dist);   // [16v x 16a] (a>=8 zero)
    }

    // ew = mask * exp(-(dist+b_s)^2); stage ew & feat tiles to per-wave LDS
#pragma unroll
    for (int j = 0; j < 8; ++j) {
      const int M   = j + hi * 8;        // vertex row within tile
      const int vtx = v0 + M;
      const float m = (vtx < nv) ? 1.0f : 0.0f;
      const float d = dist[j] + bs_n;
      const float e = (r < AQ) ? m * expf(-d * d) : 0.0f;  // zero padded cols
      ew_lds[wave][M][r] = e;
      ft_lds[wave][M][r] = feat[j] + bflr_n;
    }
    __syncthreads();

    // agg += ew^T @ feat : A[m=a,k=v] = ew[v,a], B[k=v,n=p] = feat[v,p]
#pragma unroll
    for (int kc = 0; kc < 4; ++kc) {
      const int k0 = kc * 4 + hi * 2;
      v2f aew, bft;
      aew.x = ew_lds[wave][k0 + 0][r];
      aew.y = ew_lds[wave][k0 + 1][r];
      bft.x = ft_lds[wave][k0 + 0][r];
      bft.y = ft_lds[wave][k0 + 1][r];
      agg = wmma4(aew, bft, agg);
    }
    __syncthreads();
  }

  // Reduce the 8 per-wave accumulators in LDS, then one global atomic per slot
#pragma unroll
  for (int j = 0; j < 8; ++j) {
    const int M = j + hi * 8;            // a index (rows >=8 are zero)
    atomicAdd(&agg_lds[M][r], agg[j]);
  }
  __syncthreads();

  atomicAdd(&agg_g[b * 256 + (threadIdx.x >> 4) * 16 + (threadIdx.x & 15)],
            agg_lds[threadIdx.x >> 4][threadIdx.x & 15]);
}

// ---------------------------------------------------------------------------
// Kernel 1b: AW[b,a,f] = (1/V) * sum_p agg[b,a,p] * W_out[a*P+p, f]
// a padded to 16 rows (rows 8..15 = 0). grid = B, block = 256.
// ---------------------------------------------------------------------------
__global__ __launch_bounds__(256) void garnet_aw(
    const float* __restrict__ agg_g, const float* __restrict__ W_out,
    float* __restrict__ AW)
{
  const int b   = blockIdx.x;
  const int tid = threadIdx.x;
  const int f   = tid & 15;
  if (tid < 128) {
    const int a = tid >> 4;
    float s = 0.0f;
#pragma unroll
    for (int p = 0; p < PQ; ++p)
      s += agg_g[b * 256 + a * 16 + p] * W_out[(a * PQ + p) * NFQ + f];
    AW[b * 256 + a * 16 + f] = s * (1.0f / (float)VQ);
  } else {
    const int a = ((tid - 128) >> 4) + 8;
    AW[b * 256 + a * 16 + f] = 0.0f;     // zero pad rows
  }
}

// ---------------------------------------------------------------------------
// Kernel 2: out[b,v,f] = mask * (ew[b,v,:] @ AW[b] + b_out)
// grid = (B, V/128), block = 256 (8 waves, one 16-vertex tile per wave)
// ---------------------------------------------------------------------------
__global__ __launch_bounds__(256) void garnet_out(
    const float* __restrict__ data, const int* __restrict__ num_vertex,
    const float* __restrict__ W_s,  const float* __restrict__ b_s,
    const float* __restrict__ b_out, const float* __restrict__ AW,
    float* __restrict__ out)
{
  const int b    = blockIdx.x;
  const int lane = threadIdx.x & 31;
  const int wave = threadIdx.x >> 5;
  const int r    = lane & 15;
  const int hi   = lane >> 4;
  const int t    = blockIdx.y * 8 + wave;   // tile 0..255
  const int v0   = t * 16;

  __shared__ float ew_lds[8][16][17];

  const int nv = num_vertex[b];

  v2f ws[4];
#pragma unroll
  for (int kc = 0; kc < 4; ++kc) {
    const int k0 = kc * 4 + hi * 2;
    ws[kc].x = (r < AQ) ? W_s[(k0 + 0) * AQ + r] : 0.0f;
    ws[kc].y = (r < AQ) ? W_s[(k0 + 1) * AQ + r] : 0.0f;
  }
  const float bs_n = (r < AQ) ? b_s[r] : 0.0f;

  const float* drow = data + ((size_t)b * VQ + v0 + r) * FQ;
  v2f af[4];
#pragma unroll
  for (int kc = 0; kc < 4; ++kc) {
    const int k0 = kc * 4 + hi * 2;
    af[kc].x = drow[k0 + 0];
    af[kc].y = drow[k0 + 1];
  }

  v8f dist = vzero8();
#pragma unroll
  for (int kc = 0; kc < 4; ++kc) dist = wmma4(af[kc], ws[kc], dist);

  // ew tile to LDS (C layout -> [v][a])
#pragma unroll
  for (int j = 0; j < 8; ++j) {
    const int M   = j + hi * 8;
    const int vtx = v0 + M;
    const float m = (vtx < nv) ? 1.0f : 0.0f;
    const float d = dist[j] + bs_n;
    ew_lds[wave][M][r] = (r < AQ) ? m * expf(-d * d) : 0.0f;
  }
  __syncthreads();

  // AW fragments in B-layout: B[k=a, n=f]
  const float* AWb = AW + b * 256;
  v8f o = vzero8();
#pragma unroll
  for (int kc = 0; kc < 4; ++kc) {
    const int k0 = kc * 4 + hi * 2;
    v2f aew, baw;
    aew.x = ew_lds[wave][r][k0 + 0];   // A[m=v, k=a] = ew[v,a]
    aew.y = ew_lds[wave][r][k0 + 1];
    baw.x = AWb[(k0 + 0) * 16 + r];
    baw.y = AWb[(k0 + 1) * 16 + r];
    o = wmma4(aew, baw, o);
  }

  const float bo = b_out[r];
  float* obase = out + ((size_t)b * VQ + v0) * NFQ;
#pragma unroll
  for (int j = 0; j < 8; ++j) {
    const int M   = j + hi * 8;
    const int vtx = v0 + M;
    const float m = (vtx < nv) ? 1.0f : 0.0f;
    obase[M * NFQ + r] = m * (o[j] + bo);
  }
}

// ---------------------------------------------------------------------------
extern "C" void kernel_launch(void* const* d_in, const int* in_sizes, int n_in,
                              void* d_out, int out_size, void* d_ws, size_t ws_size,
                              hipStream_t stream) {
  (void)in_sizes; (void)n_in; (void)out_size; (void)ws_size;
  const float* data       = (const float*)d_in[0];
  const int*   num_vertex = (const int*)  d_in[1];
  const float* W_flr      = (const float*)d_in[2];
  const float* b_flr      = (const float*)d_in[3];
  const float* W_s        = (const float*)d_in[4];
  const float* b_s        = (const float*)d_in[5];
  const float* W_out      = (const float*)d_in[6];
  const float* b_out      = (const float*)d_in[7];
  float* out = (float*)d_out;

  float* agg_g = (float*)d_ws;          // B*256 floats
  float* AW    = agg_g + BQ * 256;      // B*256 floats

  garnet_zero<<<(BQ * 256) / 256, 256, 0, stream>>>(agg_g, BQ * 256);
  garnet_agg<<<dim3(BQ, VQ / 256), 256, 0, stream>>>(
      data, num_vertex, W_flr, b_flr, W_s, b_s, agg_g);
  garnet_aw<<<BQ, 256, 0, stream>>>(agg_g, W_out, AW);
  garnet_out<<<dim3(BQ, VQ / 128), 256, 0, stream>>>(
      data, num_vertex, W_s, b_s, b_out, AW, out);
}